// FusionAdjacency_21320217658127
// MI455X (gfx1250) — compile-verified
//
#include <hip/hip_runtime.h>
#include <math.h>

// Problem constants from the reference (N=8192 nodes, E edges per adjacency).
static constexpr int N_NODES = 8192;

// Exact types for the gfx1250 async LDS<->global builtins, per the compiler
// diagnostic: param 0 is 'int __vector_size__(16) addrspace(1)*' (global),
// param 1 is the LDS-side pointer (addrspace(3)).
typedef int v4i __attribute__((__vector_size__(16)));
typedef __attribute__((address_space(1))) v4i gv4i;
typedef __attribute__((address_space(3))) v4i lv4i;

#if defined(__HIP_DEVICE_COMPILE__) && __has_builtin(__builtin_amdgcn_global_store_async_from_lds_b128)
#define USE_ASYNC_STORE 1
#else
#define USE_ASYNC_STORE 0
#endif

// ---------------------------------------------------------------------------
// Kernel 1: zero the dense output (256 MB) and the row-sum scratch.
// Uses the gfx1250 async LDS->global store path (ASYNCcnt-tracked,
// global_store_async_from_lds_b128): a block-wide LDS tile of zeros is
// streamed to memory 16B/lane. Falls back to plain global_store_b128.
// ---------------------------------------------------------------------------
__global__ void zero_fill_kernel(float4* __restrict__ out4,
                                 float4* __restrict__ rs4,
                                 long n_vec4, int rs_vec4) {
    long idx = (long)blockIdx.x * blockDim.x + threadIdx.x;
    float4 z = make_float4(0.f, 0.f, 0.f, 0.f);
#if USE_ASYNC_STORE
    __shared__ v4i zbuf[256];
    zbuf[threadIdx.x] = (v4i){0, 0, 0, 0};
    __syncthreads();
    if (idx < n_vec4) {
        __builtin_amdgcn_global_store_async_from_lds_b128(
            (gv4i*)((v4i*)out4 + idx),
            (lv4i*)(&zbuf[threadIdx.x]),
            /*imm offset=*/0, /*cpol=*/0);
    }
#if __has_builtin(__builtin_amdgcn_s_wait_asynccnt)
    __builtin_amdgcn_s_wait_asynccnt(0);
#endif
#else
    if (idx < n_vec4) out4[idx] = z;
#endif
    if (idx < rs_vec4) rs4[idx] = z;
}

// ---------------------------------------------------------------------------
// Kernel 2: accumulate blended row sums straight from the edge lists.
// rs[r] += alpha*vals_s[e]      (first E threads)
// rs[r] += (1-alpha)*vals_t[e]  (next E threads)
// 512K f32 atomics into a 32 KB L2-resident array.
// ---------------------------------------------------------------------------
__global__ void rowsum_kernel(const int* __restrict__ rows_s,
                              const float* __restrict__ vals_s,
                              const int* __restrict__ rows_t,
                              const float* __restrict__ vals_t,
                              const float* __restrict__ gamma,
                              float* __restrict__ rs, int E) {
    int e = blockIdx.x * blockDim.x + threadIdx.x;
    float alpha = 1.0f / (1.0f + expf(-gamma[0]));
    if (e < E) {
        atomicAdd(rs + rows_s[e], alpha * vals_s[e]);
    } else if (e < 2 * E) {
        int i = e - E;
        atomicAdd(rs + rows_t[i], (1.0f - alpha) * vals_t[i]);
    }
}

// ---------------------------------------------------------------------------
// Kernel 3: inv[r] = (rs[r]==0) ? 1 : 1/rs[r]   (reference's where(==0, 1))
// ---------------------------------------------------------------------------
__global__ void invert_rows_kernel(const float* __restrict__ rs,
                                   float* __restrict__ inv, int n) {
    int i = blockIdx.x * blockDim.x + threadIdx.x;
    if (i < n) {
        float s = rs[i];
        inv[i] = (s == 0.0f) ? 1.0f : (1.0f / s);
    }
}

// ---------------------------------------------------------------------------
// Kernel 4: scatter pre-normalized contributions into the dense output.
// Division distributes over the scatter-sum, so normalization is fused here
// and no dense read-modify-write pass over the 256 MB matrix is needed.
// ---------------------------------------------------------------------------
__global__ void scatter_kernel(const int* __restrict__ rows_s,
                               const int* __restrict__ cols_s,
                               const float* __restrict__ vals_s,
                               const int* __restrict__ rows_t,
                               const int* __restrict__ cols_t,
                               const float* __restrict__ vals_t,
                               const float* __restrict__ gamma,
                               const float* __restrict__ inv,
                               float* __restrict__ out, int E) {
    int e = blockIdx.x * blockDim.x + threadIdx.x;
    float alpha = 1.0f / (1.0f + expf(-gamma[0]));
    int r, c;
    float w;
    if (e < E) {
        r = rows_s[e];
        c = cols_s[e];
        w = alpha * vals_s[e];
    } else if (e < 2 * E) {
        int i = e - E;
        r = rows_t[i];
        c = cols_t[i];
        w = (1.0f - alpha) * vals_t[i];
    } else {
        return;
    }
    atomicAdd(out + (size_t)r * N_NODES + c, w * inv[r]);
}

extern "C" void kernel_launch(void* const* d_in, const int* in_sizes, int n_in,
                              void* d_out, int out_size, void* d_ws, size_t ws_size,
                              hipStream_t stream) {
    const int*   rows_s = (const int*)d_in[0];
    const int*   cols_s = (const int*)d_in[1];
    const float* vals_s = (const float*)d_in[2];
    const int*   rows_t = (const int*)d_in[3];
    const int*   cols_t = (const int*)d_in[4];
    const float* vals_t = (const float*)d_in[5];
    const float* gamma  = (const float*)d_in[6];
    float* out = (float*)d_out;

    const int E = in_sizes[0];

    // Workspace layout (floats): [0, N) row sums, [N, 2N) inverse row sums.
    float* rs  = (float*)d_ws;
    float* inv = rs + N_NODES;

    const int threads = 256;  // 8 wave32 waves per block

    long n_vec4  = (long)out_size / 4;   // 16,777,216 float4 stores
    int  rs_vec4 = N_NODES / 4;          // 2048 float4
    long zblocks = (n_vec4 + threads - 1) / threads;
    zero_fill_kernel<<<dim3((unsigned)zblocks), dim3(threads), 0, stream>>>(
        (float4*)out, (float4*)rs, n_vec4, rs_vec4);

    int eblocks = (2 * E + threads - 1) / threads;
    rowsum_kernel<<<dim3(eblocks), dim3(threads), 0, stream>>>(
        rows_s, vals_s, rows_t, vals_t, gamma, rs, E);

    invert_rows_kernel<<<dim3((N_NODES + threads - 1) / threads), dim3(threads), 0, stream>>>(
        rs, inv, N_NODES);

    scatter_kernel<<<dim3(eblocks), dim3(threads), 0, stream>>>(
        rows_s, cols_s, vals_s, rows_t, cols_t, vals_t, gamma, inv, out, E);
}